// PredictOutput_36464272343491
// MI455X (gfx1250) — compile-verified
//
#include <hip/hip_runtime.h>
#include <math.h>

#define IMG_WH    300.0f
#define SCALE_XYc 0.1f
#define SCALE_WHc 0.2f
#define SCORE_THR 0.5f
#define IOU_THRc  0.5f
#define MAX_OUT   100
#define TOPKN     1000
#define BATCH     64
#define M_ANC     8732
#define NCLS      21
#define NSORT     16384      // pow2 pad of 8732 for bitonic sort
#define CHUNK     4096       // LDS-resident bitonic window (32 KB of u64 keys)
#define NTILE     63         // 63*16 = 1008 >= 1000 rows
#define ADJ_ROWS  1008
#define ADJ_WORDS 32         // 1024 bits per adjacency row

typedef __attribute__((ext_vector_type(2))) float v2f;
typedef __attribute__((ext_vector_type(8))) float v8f;

#if defined(__AMDGCN__) && __has_builtin(__builtin_amdgcn_global_load_async_to_lds_b64) && \
    __has_builtin(__builtin_amdgcn_s_wait_asynccnt)
#define USE_ASYNC_LDS 1
typedef __attribute__((ext_vector_type(2))) int v2i_t;
typedef __attribute__((address_space(1))) v2i_t glb_v2i_t;   // global-AS 2xi32
typedef __attribute__((address_space(3))) v2i_t lds_v2i_t;   // LDS-AS 2xi32
#endif

// ---- workspace layout (bytes) ----
static constexpr size_t OFF_KEYS  = 0;
static constexpr size_t SZ_KEYS   = (size_t)BATCH * NSORT * sizeof(unsigned long long);
static constexpr size_t OFF_COFF  = OFF_KEYS + SZ_KEYS;                       // offset boxes (NMS space)
static constexpr size_t SZ_COFF   = (size_t)BATCH * TOPKN * 4 * sizeof(float);
static constexpr size_t OFF_CBOX  = OFF_COFF + SZ_COFF;                       // plain boxes (output)
static constexpr size_t SZ_CBOX   = SZ_COFF;
static constexpr size_t OFF_CSC   = OFF_CBOX + SZ_CBOX;
static constexpr size_t SZ_CSC    = (size_t)BATCH * TOPKN * sizeof(float);
static constexpr size_t OFF_CLB   = OFF_CSC + SZ_CSC;
static constexpr size_t SZ_CLB    = (size_t)BATCH * TOPKN * sizeof(int);
static constexpr size_t OFF_CAR   = OFF_CLB + SZ_CLB;
static constexpr size_t SZ_CAR    = (size_t)BATCH * TOPKN * sizeof(float);
static constexpr size_t OFF_ADJ   = OFF_CAR + SZ_CAR;

// order-preserving float<->u32 maps (descending-score sort key)
__device__ __forceinline__ unsigned keyFromScore(float s) {
  unsigned b = __float_as_uint(s);
  unsigned u = (b & 0x80000000u) ? ~b : (b | 0x80000000u); // ascending order
  return ~u;                                               // descending order
}
__device__ __forceinline__ float scoreFromKey(unsigned d) {
  unsigned u = ~d;
  unsigned b = (u & 0x80000000u) ? (u & 0x7FFFFFFFu) : ~u;
  return __uint_as_float(b);
}

__device__ __forceinline__ void decode_box(const float* __restrict__ bp,
                                           const float* __restrict__ anc,
                                           float out[4]) {
  float acx = anc[0], acy = anc[1], aw = anc[2], ah = anc[3];
  float x = SCALE_XYc * bp[0] * aw + acx;
  float y = SCALE_XYc * bp[1] * ah + acy;
  float w = __expf(SCALE_WHc * bp[2]) * aw;
  float h = __expf(SCALE_WHc * bp[3]) * ah;
  out[0] = fminf(fmaxf(x - 0.5f * w, 0.f), 1.f);
  out[1] = fminf(fmaxf(y - 0.5f * h, 0.f), 1.f);
  out[2] = fminf(fmaxf(x + 0.5f * w, 0.f), 1.f);
  out[3] = fminf(fmaxf(y + 0.5f * h, 0.f), 1.f);
}

// ---- K0: streaming pass. softmax(21) -> per-anchor best masked score -> sort key
__global__ void k0_keys(const float* __restrict__ bboxs_p,
                        const float* __restrict__ labels_p,
                        const float* __restrict__ ancs,
                        unsigned long long* __restrict__ keys) {
  int gid = blockIdx.x * blockDim.x + threadIdx.x;
  if (gid >= BATCH * NSORT) return;
  int b = gid / NSORT;
  int m = gid - b * NSORT;
  unsigned long long key = ~0ull;            // padding sorts last
  if (m < M_ANC) {
    const float* lp = labels_p + ((size_t)b * M_ANC + m) * NCLS;
    __builtin_prefetch(lp + NCLS * 8, 0, 1); // gfx1250 global_prefetch_b8
    float v[NCLS];
#pragma unroll
    for (int c = 0; c < NCLS; ++c) v[c] = lp[c];
    float mx = v[0];
#pragma unroll
    for (int c = 1; c < NCLS; ++c) mx = fmaxf(mx, v[c]);
    float sum = 0.f;
#pragma unroll
    for (int c = 0; c < NCLS; ++c) sum += __expf(v[c] - mx);
    int bc = 1; float bv = v[1];
#pragma unroll
    for (int c = 2; c < NCLS; ++c) { if (v[c] > bv) { bv = v[c]; bc = c; } }
    float prob = __expf(bv - mx) / sum;      // softmax prob of best non-bg class
    float box[4];
    decode_box(bboxs_p + ((size_t)b * M_ANC + m) * 4, ancs + (size_t)m * 4, box);
    float ws_ = box[2] - box[0], hs_ = box[3] - box[1];
    bool valid = (prob > SCORE_THR) && (ws_ >= 1.0f / IMG_WH) && (hs_ >= 1.0f / IMG_WH);
    float masked = valid ? prob : -1.0f;
    key = ((unsigned long long)keyFromScore(masked) << 32) |
          (unsigned)(m * (NCLS - 1) + (bc - 1));
  }
  keys[(size_t)b * NSORT + m] = key;
}

// ---- K1: per-batch bitonic sort, LDS-chunked.
// Merge distances j < CHUNK stay inside an aligned CHUNK window, so each k-stage
// stages 32 KB chunks into LDS (async global->LDS copy on gfx1250), runs all
// inner j-steps there, and writes back once. ~6x less global traffic vs naive.
__global__ void k1_sort(unsigned long long* __restrict__ keys) {
  __shared__ unsigned long long lk[CHUNK];
  unsigned long long* k = keys + (size_t)blockIdx.x * NSORT;
  for (unsigned kk = 2; kk <= NSORT; kk <<= 1) {
    unsigned j = kk >> 1;
    // cross-chunk merge distances: global compare-exchange passes
    for (; j >= CHUNK; j >>= 1) {
      for (unsigned i = threadIdx.x; i < NSORT; i += blockDim.x) {
        unsigned ixj = i ^ j;
        if (ixj > i) {
          unsigned long long a = k[i], c = k[ixj];
          bool up = ((i & kk) == 0);
          if ((a > c) == up) { k[i] = c; k[ixj] = a; }
        }
      }
      __syncthreads();
    }
    // in-chunk merge distances: fuse all remaining j-steps in LDS
    for (unsigned c0 = 0; c0 < NSORT; c0 += CHUNK) {
#ifdef USE_ASYNC_LDS
      for (unsigned i = threadIdx.x; i < CHUNK; i += blockDim.x)
        __builtin_amdgcn_global_load_async_to_lds_b64(
            (glb_v2i_t*)(k + c0 + i), (lds_v2i_t*)&lk[i], 0, 0);
      __builtin_amdgcn_s_wait_asynccnt(0);
#else
      for (unsigned i = threadIdx.x; i < CHUNK; i += blockDim.x) lk[i] = k[c0 + i];
#endif
      __syncthreads();
      for (unsigned jj = j; jj > 0; jj >>= 1) {
        for (unsigned i = threadIdx.x; i < CHUNK; i += blockDim.x) {
          unsigned ixj = i ^ jj;
          if (ixj > i) {
            unsigned long long a = lk[i], c = lk[ixj];
            bool up = (((c0 + i) & kk) == 0);
            if ((a > c) == up) { lk[i] = c; lk[ixj] = a; }
          }
        }
        __syncthreads();
      }
      for (unsigned i = threadIdx.x; i < CHUNK; i += blockDim.x) k[c0 + i] = lk[i];
      __syncthreads();
    }
  }
}

// ---- K2: gather top-1000 candidates; re-decode box, build class-offset box + area
__global__ void k2_gather(const float* __restrict__ bboxs_p,
                          const float* __restrict__ ancs,
                          const unsigned long long* __restrict__ keys,
                          float* __restrict__ coff, float* __restrict__ cbox,
                          float* __restrict__ cscore, int* __restrict__ clabel,
                          float* __restrict__ carea) {
  int r = blockIdx.x * blockDim.x + threadIdx.x;
  int b = blockIdx.y;
  if (r >= TOPKN) return;
  unsigned long long key = keys[(size_t)b * NSORT + r];
  float s = scoreFromKey((unsigned)(key >> 32));
  unsigned idx = (unsigned)key;
  size_t o1 = (size_t)b * TOPKN + r;
  bool valid = (s > SCORE_THR) && (idx < (unsigned)(M_ANC * (NCLS - 1)));
  float box[4] = {0.f, 0.f, 0.f, 0.f};
  float off[4] = {0.f, 0.f, 0.f, 0.f};
  float area = 0.f; int cls = 0; float sc = 0.f;
  if (valid) {
    int anchor = idx / (NCLS - 1);
    cls = (int)(idx % (NCLS - 1)) + 1;
    decode_box(bboxs_p + ((size_t)b * M_ANC + anchor) * 4, ancs + (size_t)anchor * 4, box);
    float ofs = 4.0f * (float)cls;
    off[0] = box[0] + ofs; off[1] = box[1] + ofs;
    off[2] = box[2] + ofs; off[3] = box[3] + ofs;
    area = (off[2] - off[0]) * (off[3] - off[1]);
    sc = s;
  }
#pragma unroll
  for (int q = 0; q < 4; ++q) { cbox[o1 * 4 + q] = box[q]; coff[o1 * 4 + q] = off[q]; }
  cscore[o1] = sc; clabel[o1] = cls; carea[o1] = area;
}

// ---- K3: adjacency bitmask via WMMA. One wave per 16-row stripe.
// union base (area_i + area_j) tile = [area,1] (16x4) x [1;area^T] (4x16) via
// V_WMMA_F32_16X16X4_F32; intersection via VALU; iou>thr compacted by wave32 ballot.
__global__ void __launch_bounds__(32) k3_adj(const float* __restrict__ coff,
                                             const float* __restrict__ carea,
                                             unsigned* __restrict__ adj) {
  int ti = blockIdx.x;          // tile row 0..62
  int b  = blockIdx.y;
  int lane = threadIdx.x;       // 0..31, wave32
  const float* off  = coff  + (size_t)b * TOPKN * 4;
  const float* area = carea + (size_t)b * TOPKN;
  unsigned* arow = adj + (size_t)b * ADJ_ROWS * ADJ_WORDS;
  __shared__ float rbx[16][4];
  __shared__ float rar[16];
  int i0 = ti * 16;
  if (lane < 16) {
    int i = i0 + lane;
    bool ok = i < TOPKN;
    rbx[lane][0] = ok ? off[i * 4 + 0] : 0.f;
    rbx[lane][1] = ok ? off[i * 4 + 1] : 0.f;
    rbx[lane][2] = ok ? off[i * 4 + 2] : 0.f;
    rbx[lane][3] = ok ? off[i * 4 + 3] : 0.f;
    rar[lane]    = ok ? area[i] : 0.f;
  }
  __syncthreads();
  // A (16x4): lanes 0-15 hold K=0,1 -> {area_m, 1}; lanes 16-31 hold K=2,3 -> 0
  v2f a;
  a.x = (lane < 16) ? rar[lane] : 0.f;
  a.y = (lane < 16) ? 1.f : 0.f;
  int cl = lane & 15;
  for (int w = 0; w < ADJ_WORDS; ++w) {
    unsigned bal0[8], bal1[8];
#pragma unroll
    for (int h = 0; h < 2; ++h) {
      int tj = 2 * w + h;
      int j  = tj * 16 + cl;
      bool jok = j < TOPKN;
      float c0 = jok ? off[j * 4 + 0] : 0.f;
      float c1 = jok ? off[j * 4 + 1] : 0.f;
      float c2 = jok ? off[j * 4 + 2] : 0.f;
      float c3 = jok ? off[j * 4 + 3] : 0.f;
      float ca = jok ? area[j] : 0.f;
      // B (4x16): lanes 0-15 hold K=0,1 -> {1, area_n}; lanes 16-31 -> 0
      v2f bv;
      bv.x = (lane < 16) ? 1.f : 0.f;
      bv.y = (lane < 16) ? ca : 0.f;
      v8f cz = {};
      v8f dmat = __builtin_amdgcn_wmma_f32_16x16x4_f32(
          false, a, false, bv, (short)0, cz, false, false);
      unsigned bal[8];
#pragma unroll
      for (int rr = 0; rr < 8; ++rr) {
        int mrow = (lane < 16) ? rr : rr + 8;     // C/D layout: lanes 16-31 -> M+8
        int i = i0 + mrow;
        float lx = fmaxf(rbx[mrow][0], c0);
        float ly = fmaxf(rbx[mrow][1], c1);
        float rx = fminf(rbx[mrow][2], c2);
        float ry = fminf(rbx[mrow][3], c3);
        float inter = fmaxf(rx - lx, 0.f) * fmaxf(ry - ly, 0.f);
        float uni = fmaxf(dmat[rr] - inter, 1e-12f);  // (area_i+area_j) - inter
        bool pred = (i < TOPKN) && (j < TOPKN) && (inter > IOU_THRc * uni);
        bal[rr] = __builtin_amdgcn_ballot_w32(pred);
      }
#pragma unroll
      for (int rr = 0; rr < 8; ++rr) { if (h == 0) bal0[rr] = bal[rr]; else bal1[rr] = bal[rr]; }
    }
    if (lane == 0) {
#pragma unroll
      for (int rr = 0; rr < 8; ++rr) {
        arow[(size_t)(i0 + rr)     * ADJ_WORDS + w] = (bal0[rr] & 0xFFFFu) | (bal1[rr] << 16);
        arow[(size_t)(i0 + rr + 8) * ADJ_WORDS + w] = (bal0[rr] >> 16) | (bal1[rr] & 0xFFFF0000u);
      }
    }
  }
}

// ---- K4: sequential greedy NMS over bitmask + output assembly
__global__ void k4_nms(const unsigned* __restrict__ adj,
                       const float* __restrict__ cbox,
                       const float* __restrict__ cscore,
                       const int* __restrict__ clabel,
                       float* __restrict__ out) {
  int b = blockIdx.x;
  const unsigned* arow = adj + (size_t)b * ADJ_ROWS * ADJ_WORDS;
  const float* box = cbox + (size_t)b * TOPKN * 4;
  const float* sc  = cscore + (size_t)b * TOPKN;
  const int*   lb  = clabel + (size_t)b * TOPKN;
  __shared__ unsigned char keep[TOPKN];
  __shared__ int sel[MAX_OUT];
  __shared__ int cnt;
  for (int i = threadIdx.x; i < TOPKN; i += blockDim.x)
    keep[i] = (sc[i] > SCORE_THR) ? 1 : 0;
  __syncthreads();
  for (int i = 0; i < TOPKN; ++i) {
    if (keep[i]) {            // uniform across block (same LDS value)
      for (int j = threadIdx.x; j < TOPKN; j += blockDim.x) {
        if (j > i && keep[j]) {
          unsigned wrd = arow[(size_t)i * ADJ_WORDS + (j >> 5)];
          if ((wrd >> (j & 31)) & 1u) keep[j] = 0;
        }
      }
    }
    __syncthreads();
  }
  if (threadIdx.x == 0) {
    int c = 0;
    for (int i = 0; i < TOPKN && c < MAX_OUT; ++i)
      if (keep[i]) sel[c++] = i;
    cnt = c;
  }
  __syncthreads();
  float* oBox = out;                                   // [B][100][4]
  float* oSc  = out + (size_t)BATCH * MAX_OUT * 4;     // [B][100]
  float* oLb  = oSc + (size_t)BATCH * MAX_OUT;         // [B][100]
  float* oV   = oLb + (size_t)BATCH * MAX_OUT;         // [B][100]
  for (int o = threadIdx.x; o < MAX_OUT; o += blockDim.x) {
    size_t oi = (size_t)b * MAX_OUT + o;
    if (o < cnt) {
      int i = sel[o];
      oBox[oi * 4 + 0] = box[i * 4 + 0];
      oBox[oi * 4 + 1] = box[i * 4 + 1];
      oBox[oi * 4 + 2] = box[i * 4 + 2];
      oBox[oi * 4 + 3] = box[i * 4 + 3];
      oSc[oi] = sc[i];
      oLb[oi] = (float)lb[i];
      oV[oi]  = 1.0f;
    } else {
      oBox[oi * 4 + 0] = 0.f; oBox[oi * 4 + 1] = 0.f;
      oBox[oi * 4 + 2] = 0.f; oBox[oi * 4 + 3] = 0.f;
      oSc[oi] = 0.f; oLb[oi] = 0.f; oV[oi] = 0.f;
    }
  }
}

extern "C" void kernel_launch(void* const* d_in, const int* in_sizes, int n_in,
                              void* d_out, int out_size, void* d_ws, size_t ws_size,
                              hipStream_t stream) {
  (void)in_sizes; (void)n_in; (void)out_size; (void)ws_size;
  const float* bboxs_p  = (const float*)d_in[0];
  const float* labels_p = (const float*)d_in[1];
  const float* ancs     = (const float*)d_in[2];
  char* ws = (char*)d_ws;
  unsigned long long* keys = (unsigned long long*)(ws + OFF_KEYS);
  float* coff   = (float*)(ws + OFF_COFF);
  float* cbox   = (float*)(ws + OFF_CBOX);
  float* cscore = (float*)(ws + OFF_CSC);
  int*   clabel = (int*)(ws + OFF_CLB);
  float* carea  = (float*)(ws + OFF_CAR);
  unsigned* adj = (unsigned*)(ws + OFF_ADJ);
  float* out = (float*)d_out;

  k0_keys<<<(BATCH * NSORT + 255) / 256, 256, 0, stream>>>(bboxs_p, labels_p, ancs, keys);
  k1_sort<<<BATCH, 1024, 0, stream>>>(keys);
  k2_gather<<<dim3((TOPKN + 255) / 256, BATCH), 256, 0, stream>>>(
      bboxs_p, ancs, keys, coff, cbox, cscore, clabel, carea);
  k3_adj<<<dim3(NTILE, BATCH), 32, 0, stream>>>(coff, carea, adj);
  k4_nms<<<BATCH, 256, 0, stream>>>(adj, cbox, cscore, clabel, out);
}